// GCN_89670327206505
// MI455X (gfx1250) — compile-verified
//
#include <hip/hip_runtime.h>
#include <hip/hip_bf16.h>

typedef float v2f __attribute__((ext_vector_type(2)));
typedef float v8f __attribute__((ext_vector_type(8)));

#define FD 64  // feature dim (16 float4 per row)

__device__ __forceinline__ void atomic_add_agent(float* p, float v) {
  __hip_atomic_fetch_add(p, v, __ATOMIC_RELAXED, __HIP_MEMORY_SCOPE_AGENT);
}

// ---------------- degree / normalization ----------------
__global__ void k_init_deg(float* __restrict__ deg, int n) {
  int i = blockIdx.x * blockDim.x + threadIdx.x;
  if (i < n) deg[i] = 1.0f;  // self-loop weight
}

__global__ void k_deg_scatter(const int* __restrict__ dst, const float* __restrict__ ew,
                              float* __restrict__ deg, int e) {
  int i = blockIdx.x * blockDim.x + threadIdx.x;
  if (i < e) atomic_add_agent(&deg[dst[i]], ew[i]);
}

__global__ void k_dinv(float* __restrict__ deg, int n) {
  int i = blockIdx.x * blockDim.x + threadIdx.x;
  if (i < n) {
    float v = deg[i];
    deg[i] = (v > 0.0f) ? rsqrtf(fmaxf(v, 1e-20f)) : 0.0f;
  }
}

// ---------------- fp32 WMMA GEMM: Y[M,64] = X[M,64] @ W[64,64] ----------------
// M must be a multiple of 16 (N=100000 is). One wave (32 lanes) per 16x16 tile;
// 4 waves/block cover the four 16-wide n-tiles of one 16-row strip.
__global__ __launch_bounds__(128)
void k_gemm64_wmma(const float* __restrict__ X, const float* __restrict__ W,
                   float* __restrict__ Y) {
  const int lane = threadIdx.x & 31;
  const int wv   = threadIdx.x >> 5;   // n-tile 0..3
  const int row  = lane & 15;          // M (A) / N (B,C) within tile
  const int kh   = lane >> 4;          // K-half select (lanes 16-31 hold K+2)
  const long long arow = ((long long)blockIdx.x * 16 + row) * FD;
  const int col = wv * 16 + row;

  v8f acc = {};
#pragma unroll
  for (int k0 = 0; k0 < FD; k0 += 4) {
    const int k = k0 + kh * 2;
    v2f a, b;
    a.x = X[arow + k];
    a.y = X[arow + k + 1];
    b.x = W[(long long)k * FD + col];
    b.y = W[(long long)(k + 1) * FD + col];
    // 8 args: (neg_a, A, neg_b, B, c_mod, C, reuse_a, reuse_b)
    acc = __builtin_amdgcn_wmma_f32_16x16x4_f32(false, a, false, b,
                                                (short)0, acc, false, false);
  }
#pragma unroll
  for (int r = 0; r < 8; ++r) {
    const int m = r + kh * 8;  // C/D layout: lanes 16-31 hold M=8..15
    Y[((long long)blockIdx.x * 16 + m) * FD + col] = acc[r];
  }
}

// ---------------- aggregation (float4 vectorized) ----------------
// out = bias + t * dinv^2   (self-loop contribution, bias folded in)
__global__ void k_agg_init(const float4* __restrict__ t4, const float4* __restrict__ bias4,
                           const float* __restrict__ dinv, float4* __restrict__ out4,
                           long long total4) {
  long long idx = (long long)blockIdx.x * blockDim.x + threadIdx.x;
  if (idx >= total4) return;
  long long i = idx >> 4;           // node
  int q = (int)(idx & 15);          // float4 chunk within row
  float di = dinv[i];
  float s = di * di;
  float4 t = t4[idx];
  float4 b = bias4[q];
  float4 o;
  o.x = b.x + t.x * s;
  o.y = b.y + t.y * s;
  o.z = b.z + t.z * s;
  o.w = b.w + t.w * s;
  out4[idx] = o;
}

// 16 threads per edge, float4 each: global_load_b128 gather + 4 fadd atomics
__global__ void k_edge_scatter(const float4* __restrict__ t4, const int* __restrict__ src,
                               const int* __restrict__ dst, const float* __restrict__ ew,
                               const float* __restrict__ dinv, float* __restrict__ out,
                               long long e16) {
  long long idx = (long long)blockIdx.x * blockDim.x + threadIdx.x;
  if (idx >= e16) return;
  long long e = idx >> 4;
  int q = (int)(idx & 15);
  int s = src[e], d = dst[e];
  float norm = dinv[s] * ew[e] * dinv[d];
  float4 v = t4[(long long)s * 16 + q];
  float* o = out + (long long)d * FD + q * 4;
  atomic_add_agent(o + 0, v.x * norm);
  atomic_add_agent(o + 1, v.y * norm);
  atomic_add_agent(o + 2, v.z * norm);
  atomic_add_agent(o + 3, v.w * norm);
}

__global__ void k_relu(const float4* __restrict__ in4, float4* __restrict__ out4,
                       long long total4) {
  long long idx = (long long)blockIdx.x * blockDim.x + threadIdx.x;
  if (idx >= total4) return;
  float4 v = in4[idx];
  v.x = fmaxf(v.x, 0.0f);
  v.y = fmaxf(v.y, 0.0f);
  v.z = fmaxf(v.z, 0.0f);
  v.w = fmaxf(v.w, 0.0f);
  out4[idx] = v;
}

// ---------------- mean pooling ----------------
__global__ void k_zero2(float* __restrict__ a, int na, float* __restrict__ b, int nb) {
  int i = blockIdx.x * blockDim.x + threadIdx.x;
  if (i < na) a[i] = 0.0f;
  if (i < nb) b[i] = 0.0f;
}

__global__ void k_pool_scatter(const float4* __restrict__ h4, const int* __restrict__ batch,
                               float* __restrict__ pooled, float* __restrict__ cnt,
                               long long total4) {
  long long idx = (long long)blockIdx.x * blockDim.x + threadIdx.x;
  if (idx >= total4) return;
  long long i = idx >> 4;
  int q = (int)(idx & 15);
  int b = batch[i];
  float4 v = h4[idx];
  float* p = pooled + (long long)b * FD + q * 4;
  atomic_add_agent(p + 0, v.x);
  atomic_add_agent(p + 1, v.y);
  atomic_add_agent(p + 2, v.z);
  atomic_add_agent(p + 3, v.w);
  if (q == 0) atomic_add_agent(&cnt[b], 1.0f);
}

__global__ void k_pool_div(float* __restrict__ pooled, const float* __restrict__ cnt,
                           int total) {
  int idx = blockIdx.x * blockDim.x + threadIdx.x;
  if (idx < total) pooled[idx] /= fmaxf(cnt[idx >> 6], 1.0f);
}

// ---------------- driver ----------------
extern "C" void kernel_launch(void* const* d_in, const int* in_sizes, int n_in,
                              void* d_out, int out_size, void* d_ws, size_t ws_size,
                              hipStream_t stream) {
  const float* x   = (const float*)d_in[0];
  const float* ew  = (const float*)d_in[1];
  const float* W1  = (const float*)d_in[2];
  const float* b1  = (const float*)d_in[3];
  const float* W2  = (const float*)d_in[4];
  const float* b2  = (const float*)d_in[5];
  const int*   ei  = (const int*)d_in[6];
  const int*   bat = (const int*)d_in[7];

  const int N = in_sizes[0] / FD;          // 100000 (multiple of 16)
  const int E = in_sizes[1];               // 1000000
  const int B = (out_size - N * FD) / FD;  // 128

  const int* src = ei;
  const int* dst = ei + E;

  float* h_out  = (float*)d_out;                 // [N,64] final node features
  float* pooled = h_out + (size_t)N * FD;        // [B,64]
  float* bufA   = (float*)d_ws;                  // [N,64] scratch
  float* dinv   = bufA + (size_t)N * FD;         // [N]
  float* cnt    = dinv + N;                      // [B]

  const long long ND4 = (long long)N * (FD / 4);   // float4 count per feature map
  const long long E16 = (long long)E * 16;         // float4 lanes per edge pass
  const int TB = 256;
  dim3 gNd4((unsigned)((ND4 + TB - 1) / TB));
  dim3 gE16((unsigned)((E16 + TB - 1) / TB));
  dim3 gN((N + TB - 1) / TB);
  dim3 gE((E + TB - 1) / TB);

  // symmetric normalization coefficients (shared by both layers)
  k_init_deg<<<gN, TB, 0, stream>>>(dinv, N);
  k_deg_scatter<<<gE, TB, 0, stream>>>(dst, ew, dinv, E);
  k_dinv<<<gN, TB, 0, stream>>>(dinv, N);

  // layer 1: t1 = x@W1 (in h_out); agg in bufA; h1 = relu(agg) -> h_out
  k_gemm64_wmma<<<N / 16, 128, 0, stream>>>(x, W1, h_out);
  k_agg_init<<<gNd4, TB, 0, stream>>>((const float4*)h_out, (const float4*)b1, dinv,
                                      (float4*)bufA, ND4);
  k_edge_scatter<<<gE16, TB, 0, stream>>>((const float4*)h_out, src, dst, ew, dinv,
                                          bufA, E16);
  k_relu<<<gNd4, TB, 0, stream>>>((const float4*)bufA, (float4*)h_out, ND4);

  // layer 2: t2 = h1@W2 (in bufA); agg in h_out; relu in place
  k_gemm64_wmma<<<N / 16, 128, 0, stream>>>(h_out, W2, bufA);
  k_agg_init<<<gNd4, TB, 0, stream>>>((const float4*)bufA, (const float4*)b2, dinv,
                                      (float4*)h_out, ND4);
  k_edge_scatter<<<gE16, TB, 0, stream>>>((const float4*)bufA, src, dst, ew, dinv,
                                          h_out, E16);
  k_relu<<<gNd4, TB, 0, stream>>>((const float4*)h_out, (float4*)h_out, ND4);

  // mean pooling over graphs
  int zb = (B * FD + TB - 1) / TB;
  k_zero2<<<zb, TB, 0, stream>>>(pooled, B * FD, cnt, B);
  k_pool_scatter<<<gNd4, TB, 0, stream>>>((const float4*)h_out, bat, pooled, cnt, ND4);
  k_pool_div<<<zb, TB, 0, stream>>>(pooled, cnt, B * FD);
}